// WeightedIDW_49426483642782
// MI455X (gfx1250) — compile-verified
//
#include <hip/hip_runtime.h>
#include <hip/hip_bf16.h>

// Weighted IDW kNN regression, fused:  dist^2 -> min-5 -> (rsqrt+normalize) @ train_y.
// WMMA f32 16x16x4 computes the cross-term tile-by-tile; the K dimension of
// the pre-scaled copies is stored permuted so each lane's WMMA operands are
// contiguous 128-bit loads. Since rsqrt(max(sq,0)+eps) is strictly decreasing
// in sq, the hot loop tracks the 5 SMALLEST squared distances (no rsqrt in the
// inner loop); rsqrt is applied to only 5 values/query in the reduce kernel.

#define NQ    4096
#define NT    16384
#define DDIM  32
#define OUTD  16
#define SEGS  8
#define SEGLEN (NT / SEGS)          // 2048 train points per segment
#define EPSV  1e-6f
#define BIGF  3.0e38f

typedef __attribute__((ext_vector_type(2))) float v2f;
typedef __attribute__((ext_vector_type(4))) float v4f;
typedef __attribute__((ext_vector_type(8))) float v8f;

// ------------------------------------------------------------- bottom-5 insert
// Keeps tv[] sorted ascending (tv[4] = largest of the 5 smallest).
__device__ __forceinline__ void bot5_insert(float (&tv)[5], int (&ti)[5],
                                            float v, int id) {
    if (v < tv[4]) {
        tv[4] = v; ti[4] = id;
#pragma unroll
        for (int k = 4; k > 0; --k) {
            if (tv[k] < tv[k - 1]) {
                float fv = tv[k]; tv[k] = tv[k - 1]; tv[k - 1] = fv;
                int   fi = ti[k]; ti[k] = ti[k - 1]; ti[k - 1] = fi;
            }
        }
    }
}

// ------------------------------------------------------------------- prep
// Scale rows by exp(-w[d]) and store with the K dimension permuted into the
// WMMA per-lane order:  pos (16h + j)  <-  K = 4*(j>>1) + (j&1) + 2h.
// A lane (half h) then reads its 16 operand floats contiguously.
__global__ __launch_bounds__(128) void idw_prep(
    const float* __restrict__ x, const float* __restrict__ tx,
    const float* __restrict__ w,
    float* __restrict__ xq_s, float* __restrict__ xt_s,
    float* __restrict__ qnorm, float* __restrict__ tnorm) {
    int i = blockIdx.x * blockDim.x + threadIdx.x;      // row id over NQ+NT
    bool isQ = i < NQ;
    const float* src = isQ ? (x + (size_t)i * DDIM) : (tx + (size_t)(i - NQ) * DDIM);
    float*       dst = isQ ? (xq_s + (size_t)i * DDIM) : (xt_s + (size_t)(i - NQ) * DDIM);
    float nrm = 0.f;
#pragma unroll
    for (int j = 0; j < DDIM; ++j) {
        int h  = j >> 4;
        int jj = j & 15;
        int K  = 4 * (jj >> 1) + (jj & 1) + 2 * h;
        float s = __expf(-w[K]);
        float v = src[K] * s;
        dst[j] = v;
        nrm   += v * v;
    }
    if (isQ) qnorm[i] = nrm; else tnorm[i - NQ] = nrm;
}

// ------------------------------------------------------------------- main
// One wave (32 lanes) per (query-tile of 16, train-segment of 2048).
// WMMA: A = 16 train rows x 32 (M,K), B = 32 x 16 queries (K,N), D = 16x16 f32.
// Lane L holds C entries for query N = L%16 and train rows r + 8*(L/16).
// Two independent 4-deep WMMA accumulation chains (c0: K 0..15, c1: K 16..31).
__global__ __launch_bounds__(32) void idw_main(
    const float* __restrict__ xq_s, const float* __restrict__ xt_s,
    const float* __restrict__ qnorm, const float* __restrict__ tnorm,
    float* __restrict__ pvals, int* __restrict__ pidx) {
    const int wid   = blockIdx.x;
    const int seg   = wid % SEGS;
    const int qtile = wid / SEGS;
    const int lane  = threadIdx.x;
    const int h     = lane >> 4;          // half selector (K split / train-row split)
    const int ql    = lane & 15;
    const int q     = qtile * 16 + ql;

    // B matrix (query tile), resident for the whole segment scan.
    const float* qrow = xq_s + (size_t)q * DDIM + 16 * h;
    v4f bq0 = *(const v4f*)(qrow + 0);
    v4f bq1 = *(const v4f*)(qrow + 4);
    v4f bq2 = *(const v4f*)(qrow + 8);
    v4f bq3 = *(const v4f*)(qrow + 12);
    v2f b[8] = {bq0.xy, bq0.zw, bq1.xy, bq1.zw, bq2.xy, bq2.zw, bq3.xy, bq3.zw};
    const float qn = qnorm[q];

    float tv[5] = {BIGF, BIGF, BIGF, BIGF, BIGF};
    int   ti[5] = {0, 0, 0, 0, 0};

    const int tbeg = seg * SEGLEN;
#pragma unroll 2
    for (int t0 = tbeg; t0 < tbeg + SEGLEN; t0 += 16) {
        // A matrix: train rows t0..t0+15; lane's 16 floats are contiguous.
        const float* arow = xt_s + (size_t)(t0 + ql) * DDIM + 16 * h;
        v4f a0 = *(const v4f*)(arow + 0);
        v4f a1 = *(const v4f*)(arow + 4);
        v4f a2 = *(const v4f*)(arow + 8);
        v4f a3 = *(const v4f*)(arow + 12);

        v8f c0 = {0.f, 0.f, 0.f, 0.f, 0.f, 0.f, 0.f, 0.f};
        v8f c1 = c0;
        c0 = __builtin_amdgcn_wmma_f32_16x16x4_f32(false, a0.xy, false, b[0], (short)0, c0, false, false);
        c1 = __builtin_amdgcn_wmma_f32_16x16x4_f32(false, a2.xy, false, b[4], (short)0, c1, false, false);
        c0 = __builtin_amdgcn_wmma_f32_16x16x4_f32(false, a0.zw, false, b[1], (short)0, c0, false, false);
        c1 = __builtin_amdgcn_wmma_f32_16x16x4_f32(false, a2.zw, false, b[5], (short)0, c1, false, false);
        c0 = __builtin_amdgcn_wmma_f32_16x16x4_f32(false, a1.xy, false, b[2], (short)0, c0, false, false);
        c1 = __builtin_amdgcn_wmma_f32_16x16x4_f32(false, a3.xy, false, b[6], (short)0, c1, false, false);
        c0 = __builtin_amdgcn_wmma_f32_16x16x4_f32(false, a1.zw, false, b[3], (short)0, c0, false, false);
        c1 = __builtin_amdgcn_wmma_f32_16x16x4_f32(false, a3.zw, false, b[7], (short)0, c1, false, false);

        // Train norms for this lane's 8 rows: tnorm[t0 + 8h + r]
        const float* tnp = tnorm + t0 + 8 * h;
        v4f tna = *(const v4f*)(tnp);
        v4f tnb = *(const v4f*)(tnp + 4);
        float tn[8] = {tna.x, tna.y, tna.z, tna.w, tnb.x, tnb.y, tnb.z, tnb.w};

        // Track smallest squared distances (monotonic proxy for largest dinv).
#pragma unroll
        for (int r = 0; r < 8; ++r) {
            float sq = qn + tn[r];
            sq = fmaf(-2.0f, c0[r], sq);
            sq = fmaf(-2.0f, c1[r], sq);
            bot5_insert(tv, ti, sq, t0 + 8 * h + r);
        }
    }

    // Merge lane L with lane L^16 (other train-row half, same query).
    float pv[5]; int pi[5];
#pragma unroll
    for (int k = 0; k < 5; ++k) {
        pv[k] = __shfl_xor(tv[k], 16, 32);
        pi[k] = __shfl_xor(ti[k], 16, 32);
    }
#pragma unroll
    for (int k = 0; k < 5; ++k) bot5_insert(tv, ti, pv[k], pi[k]);

    if (lane < 16) {
        int base = (q * SEGS + seg) * 5;
#pragma unroll
        for (int k = 0; k < 5; ++k) { pvals[base + k] = tv[k]; pidx[base + k] = ti[k]; }
    }
}

// ------------------------------------------------------------------- reduce
// Merge SEGS partial min-5 lists per query, convert sq -> dinv, normalize,
// and emit out = w @ train_y for the 5 selected neighbors.
__global__ __launch_bounds__(128) void idw_reduce(
    const float* __restrict__ pvals, const int* __restrict__ pidx,
    const float* __restrict__ train_y, float* __restrict__ out) {
    int q = blockIdx.x * blockDim.x + threadIdx.x;
    if (q >= NQ) return;

    float tv[5] = {BIGF, BIGF, BIGF, BIGF, BIGF};
    int   ti[5] = {0, 0, 0, 0, 0};
    for (int s = 0; s < SEGS; ++s) {
        int base = (q * SEGS + s) * 5;
#pragma unroll
        for (int k = 0; k < 5; ++k)
            bot5_insert(tv, ti, pvals[base + k], pidx[base + k]);
    }

    // dinv = 1/sqrt(max(sq,0)+eps)  (exactly 5 per query)
    float wv[5];
#pragma unroll
    for (int k = 0; k < 5; ++k)
        wv[k] = rsqrtf(fmaxf(tv[k], 0.f) + EPSV);

    float sw  = wv[0] + wv[1] + wv[2] + wv[3] + wv[4];
    float inv = 1.f / sw;

    v4f acc0 = {0.f, 0.f, 0.f, 0.f}, acc1 = acc0, acc2 = acc0, acc3 = acc0;
#pragma unroll
    for (int k = 0; k < 5; ++k) {
        const float* yr = train_y + (size_t)ti[k] * OUTD;
        float wgt = wv[k];
        v4f y0 = *(const v4f*)(yr + 0);
        v4f y1 = *(const v4f*)(yr + 4);
        v4f y2 = *(const v4f*)(yr + 8);
        v4f y3 = *(const v4f*)(yr + 12);
        acc0 += wgt * y0; acc1 += wgt * y1; acc2 += wgt * y2; acc3 += wgt * y3;
    }
    float* op = out + (size_t)q * OUTD;
    *(v4f*)(op + 0)  = acc0 * inv;
    *(v4f*)(op + 4)  = acc1 * inv;
    *(v4f*)(op + 8)  = acc2 * inv;
    *(v4f*)(op + 12) = acc3 * inv;
}

// ------------------------------------------------------------------- launch
extern "C" void kernel_launch(void* const* d_in, const int* in_sizes, int n_in,
                              void* d_out, int out_size, void* d_ws, size_t ws_size,
                              hipStream_t stream) {
    const float* x       = (const float*)d_in[0];   // (4096, 32)
    const float* train_x = (const float*)d_in[1];   // (16384, 32)
    const float* train_y = (const float*)d_in[2];   // (16384, 16)
    const float* weights = (const float*)d_in[3];   // (32,)
    float* out = (float*)d_out;                     // (4096, 16)

    // Workspace layout (floats):
    float* ws    = (float*)d_ws;
    float* xq_s  = ws;                          // 4096*32   = 131072
    float* xt_s  = ws + 131072;                 // 16384*32  = 524288
    float* qnorm = ws + 655360;                 // 4096
    float* tnorm = ws + 659456;                 // 16384
    float* pvals = ws + 675840;                 // 4096*SEGS*5 = 163840
    int*   pidx  = (int*)(ws + 839680);         // 163840 ints
    // total ~4.0 MB

    idw_prep<<<(NQ + NT) / 128, 128, 0, stream>>>(x, train_x, weights,
                                                  xq_s, xt_s, qnorm, tnorm);
    idw_main<<<(NQ / 16) * SEGS, 32, 0, stream>>>(xq_s, xt_s, qnorm, tnorm,
                                                  pvals, pidx);
    idw_reduce<<<NQ / 128, 128, 0, stream>>>(pvals, pidx, train_y, out);
}